// PagedAttention_64957085385466
// MI455X (gfx1250) — compile-verified
//
#include <hip/hip_runtime.h>

typedef float v2f __attribute__((ext_vector_type(2)));
typedef float v8f __attribute__((ext_vector_type(8)));

#define NSEQ 64
#define NHEAD 32
#define NKV 8
#define QPK 4
#define HS 128
#define BSZ 16
#define MAXB 128
#define NWAVE 8
#define QK_SCALE 0.08838834764831845f
#define NEG_BIG (-1.0e30f)

// One workgroup per (seq, kv_head); 8 wave32s flash-decode the context blocks.
__global__ __launch_bounds__(256)
void paged_attn_wmma(const float* __restrict__ query,
                     const float* __restrict__ knew,
                     const float* __restrict__ vnew,
                     const float* __restrict__ kcache,
                     const float* __restrict__ vcache,
                     const int*   __restrict__ btab,
                     const int*   __restrict__ ctxlens,
                     const float* __restrict__ alibi,
                     float*       __restrict__ out)
{
  const int kv   = blockIdx.x;
  const int seq  = blockIdx.y;
  const int tid  = threadIdx.x;
  const int wave = tid >> 5;
  const int lane = tid & 31;
  const int ln   = lane & 15;
  const bool lo  = lane < 16;

  __shared__ float qs[QPK * HS];              // Q pre-scaled by 1/sqrt(HS)
  __shared__ float acc_s[NWAVE][QPK][HS];     // per-wave unnormalized PV partials
  __shared__ float m_s[NWAVE][QPK];
  __shared__ float l_s[NWAVE][QPK];
  __shared__ float sl_s[QPK];                 // new-token logits

  // ---- stage Q (pre-scaled) into LDS ----
  #pragma unroll
  for (int k = 0; k < 2; ++k) {
    int idx = tid + 256 * k;
    int h = idx >> 7, d = idx & 127;
    qs[idx] = query[(size_t)seq * (NHEAD * HS) + (kv * QPK + h) * HS + d] * QK_SCALE;
  }
  __syncthreads();

  const int ctx = ctxlens[seq];
  const int nb  = (ctx + 14) >> 4;            // ceil((ctx-1)/16): cached tokens only

  const float slope = (ln < QPK) ? alibi[kv * QPK + ln] : 0.0f;

  float m_run = NEG_BIG, l_run = 0.0f;
  v8f accv[8] = {};                           // D[m=head(4 live), n=hs] per hs tile

  const float* qlane = qs + ln * HS + (lo ? 0 : 2);

  for (int b = wave; b < nb; b += NWAVE) {
    const int phys = btab[seq * MAXB + b];
    const float* kb = kcache + ((size_t)phys * NKV + kv) * (HS * BSZ);
    const float* vb = vcache + ((size_t)phys * NKV + kv) * (HS * BSZ);
    if (b + NWAVE < nb) {
      int pn = btab[seq * MAXB + b + NWAVE];
      __builtin_prefetch(kcache + ((size_t)pn * NKV + kv) * (HS * BSZ), 0, 0);
    }

    // ---- scores[16 tok x 16 hd] = K_tile(16x128) x Q^T : 32x WMMA f32 16x16x4 ----
    // key_cache tile layout: [hs_chunk(16)][tok(16)][x(8)], hs = chunk*8 + x.
    // A frag (tok-major): lane<16 holds K[tok=ln][4i+0,4i+1]; lane>=16 holds [4i+2,4i+3].
    const float* klane = kb + ln * 8 + (lo ? 0 : 2);
    v8f sc0 = {}, sc1 = {};                   // 2 interleaved chains for ILP
    #pragma unroll
    for (int i = 0; i < 32; i += 2) {
      v2f a0 = *(const v2f*)(klane + (i >> 1) * 128);
      v2f a1 = *(const v2f*)(klane + (i >> 1) * 128 + 4);
      v2f b0 = {}, b1 = {};
      if (ln < QPK) {                          // dead head columns stay 0
        b0 = *(const v2f*)(qlane + 4 * i);
        b1 = *(const v2f*)(qlane + 4 * i + 4);
      }
      sc0 = __builtin_amdgcn_wmma_f32_16x16x4_f32(false, a0, false, b0, (short)0,
                                                  sc0, false, false);
      sc1 = __builtin_amdgcn_wmma_f32_16x16x4_f32(false, a1, false, b1, (short)0,
                                                  sc1, false, false);
    }
    v8f sc = sc0 + sc1;

    // ---- ALiBi + causal mask + online softmax (head = ln, tok = r + 8*hi) ----
    float p[8];
    float mb = NEG_BIG;
    const int pos0 = b * BSZ + (lo ? 0 : 8);
    #pragma unroll
    for (int r = 0; r < 8; ++r) {
      int pos = pos0 + r;
      float v = sc[r] + slope * (float)(pos - ctx + 1);
      p[r] = (pos < ctx - 1) ? v : NEG_BIG;   // new token handled separately
      mb = fmaxf(mb, p[r]);
    }
    mb = fmaxf(mb, __shfl_xor(mb, 16, 32));
    const float m_new = fmaxf(m_run, mb);
    const float scl = __expf(m_run - m_new);
    float psum = 0.0f;
    #pragma unroll
    for (int r = 0; r < 8; ++r) { p[r] = __expf(p[r] - m_new); psum += p[r]; }
    psum += __shfl_xor(psum, 16, 32);
    l_run = l_run * scl + psum;
    m_run = m_new;

    // rescale PV accumulators: row m of D is head m -> broadcast lane m's factor
    const float s0 = __shfl(scl, 0, 32);
    const float s1 = __shfl(scl, 1, 32);
    const float s2 = __shfl(scl, 2, 32);
    const float s3 = __shfl(scl, 3, 32);
    #pragma unroll
    for (int t = 0; t < 8; ++t) {
      accv[t][0] *= s0; accv[t][1] *= s1; accv[t][2] *= s2; accv[t][3] *= s3;
    }

    // ---- out(4x128) += P(4x16) x V_tile(16x128): 32x WMMA f32 16x16x4 ----
    float tq[8];
    #pragma unroll
    for (int r = 0; r < 8; ++r) tq[r] = __shfl_xor(p[r], 16, 32);

    #pragma unroll
    for (int j = 0; j < 4; ++j) {              // K-dim = tokens 4j..4j+3
      const int o = (j & 1) * 4;
      v2f a;                                   // A frag: row = head, K split by lane half
      if (j < 2) { a[0] = lo ? p[o]    : tq[o + 2]; a[1] = lo ? p[o + 1]  : tq[o + 3]; }
      else       { a[0] = lo ? tq[o]   : p[o + 2];  a[1] = lo ? tq[o + 1] : p[o + 3];  }
      // value_cache tile layout: [hs(128)][tok(16)]
      const float* vl = vb + ln * BSZ + 4 * j + (lo ? 0 : 2);
      #pragma unroll
      for (int t = 0; t < 8; ++t) {            // 8 independent accumulator chains
        v2f bv = *(const v2f*)(vl + t * 256);
        accv[t] = __builtin_amdgcn_wmma_f32_16x16x4_f32(false, a, false, bv, (short)0,
                                                        accv[t], false, false);
      }
    }
  }

  // ---- write per-wave partials ----
  if (lane < QPK) { m_s[wave][lane] = m_run; l_s[wave][lane] = l_run; }
  if (lo) {
    #pragma unroll
    for (int t = 0; t < 8; ++t)
      #pragma unroll
      for (int r = 0; r < QPK; ++r)
        acc_s[wave][r][t * 16 + ln] = accv[t][r];
  }
  // new-token logit: s_last[h] = (Q*SCALE) . k_new  (dist = 0 => no ALiBi term)
  if (tid < QPK) {
    const float* kn = knew + (size_t)seq * (NKV * HS) + kv * HS;
    float d = 0.0f;
    for (int i = 0; i < HS; ++i) d += qs[tid * HS + i] * kn[i];
    sl_s[tid] = d;
  }
  __syncthreads();

  // ---- combine waves + new token, normalize, store ----
  const float* vn = vnew + (size_t)seq * (NKV * HS) + kv * HS;
  #pragma unroll
  for (int k = 0; k < 2; ++k) {
    int idx = tid + 256 * k;
    int h = idx >> 7, d = idx & 127;
    float sl = sl_s[h];
    float M = sl;
    #pragma unroll
    for (int w = 0; w < NWAVE; ++w) M = fmaxf(M, m_s[w][h]);
    float el  = __expf(sl - M);
    float tot = el;
    float num = el * vn[d];
    #pragma unroll
    for (int w = 0; w < NWAVE; ++w) {
      float e = __expf(m_s[w][h] - M);
      tot += l_s[w][h] * e;
      num += acc_s[w][h][d] * e;
    }
    out[(size_t)seq * (NHEAD * HS) + (kv * QPK + h) * HS + d] = num / tot;
  }
}

extern "C" void kernel_launch(void* const* d_in, const int* in_sizes, int n_in,
                              void* d_out, int out_size, void* d_ws, size_t ws_size,
                              hipStream_t stream) {
  const float* query  = (const float*)d_in[0];
  const float* knew   = (const float*)d_in[1];
  const float* vnew   = (const float*)d_in[2];
  const float* kcache = (const float*)d_in[3];
  const float* vcache = (const float*)d_in[4];
  const int*   btab   = (const int*)d_in[5];
  const int*   ctx    = (const int*)d_in[6];
  // d_in[7] = slot_mapping: implied by context_lens (last slot = ctx-1); unused.
  const float* alibi  = (const float*)d_in[8];

  dim3 grid(NKV, NSEQ);
  dim3 block(256);
  paged_attn_wmma<<<grid, block, 0, stream>>>(query, knew, vnew, kcache, vcache,
                                              btab, ctx, alibi, (float*)d_out);
}